// GraphThreshold_68367289418039
// MI455X (gfx1250) — compile-verified
//
#include <hip/hip_runtime.h>
#include <cstdint>

#define NNODES 40000
#define LDSCAP 2048
#define SCAN_THREADS 1024
#define SORT_THREADS 64

// ---------------------------------------------------------------- utilities

__global__ void gt_zero_i32(int* __restrict__ p, int n) {
  int i = blockIdx.x * blockDim.x + threadIdx.x;
  if (i < n) p[i] = 0;
}

__device__ __forceinline__ void gt_load_edge(const int* __restrict__ ei,
                                             const float* __restrict__ ea,
                                             int E, int i,
                                             int& r, int& c, float& a) {
  if (i < E) {            // original direction (row, col)
    r = ei[i];
    c = ei[E + i];
    a = ea[i];
  } else {                // symmetrized direction (col, row)
    int j = i - E;
    r = ei[E + j];
    c = ei[j];
    a = ea[j];
  }
}

// -------------------------------------------------- pass 1: per-row histogram

__global__ void gt_count(const int* __restrict__ ei, const float* __restrict__ ea,
                         const float* __restrict__ t, int E,
                         int* __restrict__ counts) {
  int i = blockIdx.x * blockDim.x + threadIdx.x;
  int E2 = 2 * E;
  if (i >= E2) return;
  int r, c; float a;
  gt_load_edge(ei, ea, E, i, r, c, a);
  if (a <= t[0]) atomicAdd(&counts[r], 1);
}

// ---------------------------- single-workgroup exclusive scan over n counters
// offsets[0..n] written (offsets[n] = total); optional atomic-cursor copy and
// optional float total written to totalOut[totalIdx].

__global__ void __launch_bounds__(SCAN_THREADS)
gt_scan(const int* __restrict__ counts, int n,
        int* __restrict__ offsets, int* __restrict__ cursor,
        float* __restrict__ totalOut, int totalIdx) {
  __shared__ int s[SCAN_THREADS];
  int tid = threadIdx.x;
  int chunk = (n + SCAN_THREADS - 1) / SCAN_THREADS;
  int beg = tid * chunk;
  int end = beg + chunk; if (end > n) end = n;

  int sum = 0;
  for (int i = beg; i < end; ++i) sum += counts[i];
  s[tid] = sum;
  __syncthreads();
  for (int off = 1; off < SCAN_THREADS; off <<= 1) {
    int v = (tid >= off) ? s[tid - off] : 0;
    __syncthreads();
    s[tid] += v;
    __syncthreads();
  }
  int running = s[tid] - sum;              // exclusive prefix of this chunk
  for (int i = beg; i < end; ++i) {
    offsets[i] = running;
    if (cursor) cursor[i] = running;
    running += counts[i];
  }
  if (tid == SCAN_THREADS - 1) {
    int total = s[SCAN_THREADS - 1];
    offsets[n] = total;
    if (totalOut) totalOut[totalIdx] = (float)total;
  }
}

// -------------------------------------------------- pass 2: bucketed scatter

__global__ void gt_scatter(const int* __restrict__ ei, const float* __restrict__ ea,
                           const float* __restrict__ t, int E,
                           int* __restrict__ cursor,
                           int* __restrict__ bcol, float* __restrict__ bval) {
  int i = blockIdx.x * blockDim.x + threadIdx.x;
  int E2 = 2 * E;
  if (i >= E2) return;
  int r, c; float a;
  gt_load_edge(ei, ea, E, i, r, c, a);
  if (a <= t[0]) {
    int pos = atomicAdd(&cursor[r], 1);
    bcol[pos] = c;
    bval[pos] = a;
  }
}

// ------------------- per-row LDS bitonic sort by (col, value-bits) + merge.
// Secondary value key makes duplicate summation order deterministic even
// though the atomic scatter order is not. Mean bucket len = 64, so a
// 64-thread (2-wave) block matches the typical bitonic width; strided loops
// still handle the (statistically absent) pad=2048 worst case.

__global__ void __launch_bounds__(SORT_THREADS)
gt_sortmerge(const int* __restrict__ offsets,
             int* __restrict__ bcol, float* __restrict__ bval,
             int* __restrict__ uniqueCount) {
  __shared__ int   sc[LDSCAP];
  __shared__ float sv[LDSCAP];
  int b    = blockIdx.x;
  int tid  = threadIdx.x;
  int base = offsets[b];
  int len  = offsets[b + 1] - base;

  if (len <= 1) {
    if (tid == 0) uniqueCount[b] = len;   // 0 or 1 elems already "merged"
    return;
  }

  if (len > LDSCAP) {                     // statistically never hit (mean ~64)
    if (tid == 0) {
      for (int i = 1; i < len; ++i) {     // serial insertion sort in global
        int ck = bcol[base + i]; float vv = bval[base + i];
        int j = i - 1;
        while (j >= 0 && (bcol[base + j] > ck ||
                          (bcol[base + j] == ck && bval[base + j] > vv))) {
          bcol[base + j + 1] = bcol[base + j];
          bval[base + j + 1] = bval[base + j];
          --j;
        }
        bcol[base + j + 1] = ck; bval[base + j + 1] = vv;
      }
      int u = 0; int curc = bcol[base]; float acc = bval[base];
      for (int i = 1; i < len; ++i) {
        int ck = bcol[base + i]; float vv = bval[base + i];
        if (ck == curc) { acc += vv; }
        else { bcol[base + u] = curc; bval[base + u] = acc; ++u; curc = ck; acc = vv; }
      }
      bcol[base + u] = curc; bval[base + u] = acc; ++u;
      uniqueCount[b] = u;
    }
    return;
  }

  int pad = 2; while (pad < len) pad <<= 1;          // power-of-two size
  for (int i = tid; i < pad; i += SORT_THREADS) {
    if (i < len) { sc[i] = bcol[base + i]; sv[i] = bval[base + i]; }
    else         { sc[i] = 0x7fffffff;     sv[i] = 0.0f; }
  }
  __syncthreads();

  for (int k = 2; k <= pad; k <<= 1) {
    for (int j = k >> 1; j > 0; j >>= 1) {
      for (int i = tid; i < pad; i += SORT_THREADS) {
        int ixj = i ^ j;
        if (ixj > i) {                     // disjoint pairs within a substep
          int   ci = sc[i], cj = sc[ixj];
          float vi = sv[i], vj = sv[ixj];
          bool gt = (ci > cj) || (ci == cj && vi > vj);
          bool up = ((i & k) == 0);
          if (gt == up) { sc[i] = cj; sc[ixj] = ci; sv[i] = vj; sv[ixj] = vi; }
        }
      }
      __syncthreads();
    }
  }

  if (tid == 0) {                         // serial run-merge (~len iters)
    int u = 0; int curc = sc[0]; float acc = sv[0];
    for (int i = 1; i < len; ++i) {
      if (sc[i] == curc) { acc += sv[i]; }
      else { bcol[base + u] = curc; bval[base + u] = acc; ++u; curc = sc[i]; acc = sv[i]; }
    }
    bcol[base + u] = curc; bval[base + u] = acc; ++u;
    uniqueCount[b] = u;
  }
}

// ------------------------------------------------ output: padding + compact
// Fill only the invalid tail [total, E2): the valid prefix is fully written
// by gt_compact, so skipping it saves ~half the output store traffic.

__global__ void gt_fill_tail(const int* __restrict__ uoff,
                             float* __restrict__ out, int E2) {
  int total = uoff[NNODES];
  int i = blockIdx.x * blockDim.x + threadIdx.x;
  if (i < total || i >= E2) return;
  out[i]          = -1.0f;   // row padding
  out[E2 + i]     = -1.0f;   // col padding
  out[2 * E2 + i] =  0.0f;   // attr padding
}

__global__ void gt_compact(const int* __restrict__ offsets, const int* __restrict__ uoff,
                           const int* __restrict__ bcol, const float* __restrict__ bval,
                           const int* __restrict__ uniqueCount,
                           float* __restrict__ out, int E2) {
  int b   = blockIdx.x;
  int u   = uniqueCount[b];
  int src = offsets[b];
  int dst = uoff[b];
  for (int j = threadIdx.x; j < u; j += blockDim.x) {
    int p = dst + j;
    out[p]          = (float)b;                 // row
    out[E2 + p]     = (float)bcol[src + j];     // col
    out[2 * E2 + p] = bval[src + j];            // summed attr
  }
}

// ---------------------------------------------------------------- launcher

static inline size_t gt_align256(size_t x) { return (x + 255) & ~size_t(255); }

extern "C" void kernel_launch(void* const* d_in, const int* in_sizes, int n_in,
                              void* d_out, int out_size, void* d_ws, size_t ws_size,
                              hipStream_t stream) {
  const int*   ei = (const int*)d_in[0];     // [2, E] flat: rows then cols
  const float* ea = (const float*)d_in[1];   // [E]
  const float* t  = (const float*)d_in[2];   // [1]
  const int E  = in_sizes[1];
  const int E2 = 2 * E;
  float* out = (float*)d_out;                // [2E rows | 2E cols | 2E attr | 1 count]

  char* w = (char*)d_ws;
  int*   counts      = (int*)w;  w += gt_align256(sizeof(int)   * NNODES);
  int*   offsets     = (int*)w;  w += gt_align256(sizeof(int)   * (NNODES + 1));
  int*   cursor      = (int*)w;  w += gt_align256(sizeof(int)   * NNODES);
  int*   uniqueCount = (int*)w;  w += gt_align256(sizeof(int)   * NNODES);
  int*   uoff        = (int*)w;  w += gt_align256(sizeof(int)   * (NNODES + 1));
  int*   bcol        = (int*)w;  w += gt_align256(sizeof(int)   * (size_t)E2);
  float* bval        = (float*)w;

  const int T = 256;
  const int gE2 = (E2 + T - 1) / T;
  const int gNB = (NNODES + T - 1) / T;

  gt_zero_i32 <<<gNB, T, 0, stream>>>(counts, NNODES);
  gt_count    <<<gE2, T, 0, stream>>>(ei, ea, t, E, counts);
  gt_scan     <<<1, SCAN_THREADS, 0, stream>>>(counts, NNODES, offsets, cursor,
                                               nullptr, 0);
  gt_scatter  <<<gE2, T, 0, stream>>>(ei, ea, t, E, cursor, bcol, bval);
  gt_sortmerge<<<NNODES, SORT_THREADS, 0, stream>>>(offsets, bcol, bval, uniqueCount);
  gt_scan     <<<1, SCAN_THREADS, 0, stream>>>(uniqueCount, NNODES, uoff,
                                               nullptr, out, 3 * E2);
  gt_fill_tail<<<gE2, T, 0, stream>>>(uoff, out, E2);
  gt_compact  <<<NNODES, T, 0, stream>>>(offsets, uoff, bcol, bval,
                                         uniqueCount, out, E2);
}